// Decoder_11476152615361
// MI455X (gfx1250) — compile-verified
//
#include <hip/hip_runtime.h>
#include <hip/hip_bf16.h>
#include <math.h>

typedef __attribute__((ext_vector_type(2))) float v2f;
typedef __attribute__((ext_vector_type(8))) float v8f;

#define HDIM 2048
#define VDIM 50257
#define LDIM 80

// ---------------- workspace layout (float offsets) ----------------
#define OFF_CAT     0        // 2H = 4096   [h0 || embedded]
#define OFF_ALOG    4096     // 80 (pad 128) attn logits
#define OFF_AW      4224     // 80 (pad 128) attn weights
#define OFF_COMB    4352     // 2H = 4096   [embedded || attn_applied]
#define OFF_LSTMIN  8448     // H
#define OFF_GATES   10496    // 4H = 8192
#define OFF_HNEW    18688    // H
#define OFF_CNEW    20736    // H
#define OFF_LOGITS  22784    // V (pad 50304)
#define OFF_PM      73088    // 128 partial maxima
#define OFF_PS      73216    // 128 partial sums
#define OFF_STATS   73344    // {max, log(sumexp)}

// ------------------------------------------------------------------
// Generic y = act( W(rows x cols) * x + bias [+ y] ) via WMMA f32 16x16x4.
// One wave handles a 16-row tile; 8 tiles per 256-thread block.
// x is staged in LDS; W is streamed with non-temporal loads.
// ------------------------------------------------------------------
__global__ __launch_bounds__(256) void wmma_gemv_kernel(
    const float* __restrict__ W, const float* __restrict__ x,
    const float* __restrict__ bias, float* __restrict__ y,
    int rows, int cols, int accumulate, int relu)
{
    extern __shared__ float xs[];
    for (int i = threadIdx.x; i < cols; i += blockDim.x) xs[i] = x[i];
    __syncthreads();

    const int wave = threadIdx.x >> 5;
    const int lane = threadIdx.x & 31;
    const int half = lane >> 4;        // 0: K slots {0,1}, 1: K slots {2,3}
    const int m    = lane & 15;        // row within tile
    const int v0   = (blockIdx.x * 8 + wave) * 16;
    if (v0 >= rows) return;

    if (v0 + 16 <= rows) {
        // --- full 16-row tile: WMMA path (whole wave active, EXEC all-1s) ---
        const float* Arow = W + (size_t)(v0 + m) * (size_t)cols + 2 * half;
        v8f acc = {};
        #pragma unroll 8
        for (int k = 0; k < cols; k += 4) {
            // A tile 16x4: lane(half=0) vgpr0=W[r,k],   vgpr1=W[r,k+1]
            //              lane(half=1) vgpr0=W[r,k+2], vgpr1=W[r,k+3]
            v2f a = __builtin_nontemporal_load((const v2f*)(Arow + k));
            // B 4x16 with x-chunk replicated across all 16 columns
            v2f b;
            b[0] = xs[k + 2 * half];
            b[1] = xs[k + 2 * half + 1];
            acc = __builtin_amdgcn_wmma_f32_16x16x4_f32(
                false, a, false, b, (short)0, acc, false, false);
        }
        // D: vgpr r, lanes0-15 -> M=r, lanes16-31 -> M=8+r (all columns equal)
        if (m == 0) {
            int base = v0 + 8 * half;
            #pragma unroll
            for (int r = 0; r < 8; ++r) {
                float val = acc[r] + bias[base + r];
                if (accumulate) val += y[base + r];
                if (relu) val = fmaxf(val, 0.0f);
                y[base + r] = val;
            }
        }
    } else {
        // --- tail tile (< 16 rows): scalar dot + wave reduction ---
        for (int row = v0; row < rows; ++row) {
            const float* Wr = W + (size_t)row * (size_t)cols;
            float s = 0.0f;
            for (int k = lane; k < cols; k += 32) s += Wr[k] * xs[k];
            #pragma unroll
            for (int off = 16; off > 0; off >>= 1) s += __shfl_down(s, off, 32);
            if (lane == 0) {
                float val = s + bias[row];
                if (accumulate) val += y[row];
                if (relu) val = fmaxf(val, 0.0f);
                y[row] = val;
            }
        }
    }
}

// cat_he = [h0 || emb_W[x]]
__global__ void prep_cat_kernel(const int* __restrict__ x,
                                const float* __restrict__ h,
                                const float* __restrict__ emb_W,
                                float* __restrict__ cat_he)
{
    int i = blockIdx.x * blockDim.x + threadIdx.x;
    if (i >= 2 * HDIM) return;
    const float* emb = emb_W + (size_t)x[0] * HDIM;
    cat_he[i] = (i < HDIM) ? h[i] : emb[i - HDIM];
}

// softmax over L=80; also writes attention output (d_out + V)
__global__ void attn_softmax_kernel(const float* __restrict__ logits,
                                    float* __restrict__ attn_w,
                                    float* __restrict__ out_attn)
{
    __shared__ float sh[128];
    int t = threadIdx.x;
    float v = (t < LDIM) ? logits[t] : -INFINITY;
    sh[t] = v; __syncthreads();
    for (int s = 64; s > 0; s >>= 1) { if (t < s) sh[t] = fmaxf(sh[t], sh[t + s]); __syncthreads(); }
    float mx = sh[0]; __syncthreads();
    float e = (t < LDIM) ? expf(v - mx) : 0.0f;
    sh[t] = e; __syncthreads();
    for (int s = 64; s > 0; s >>= 1) { if (t < s) sh[t] += sh[t + s]; __syncthreads(); }
    float sum = sh[0];
    if (t < LDIM) {
        float w = e / sum;
        attn_w[t]  = w;
        out_attn[t] = w;
    }
}

// comb = [emb || attn_w @ enc]
__global__ void attn_apply_kernel(const float* __restrict__ attn_w,
                                  const float* __restrict__ enc,
                                  const int* __restrict__ x,
                                  const float* __restrict__ emb_W,
                                  float* __restrict__ comb)
{
    int j = blockIdx.x * blockDim.x + threadIdx.x;
    if (j >= HDIM) return;
    const float* emb = emb_W + (size_t)x[0] * HDIM;
    float s = 0.0f;
    #pragma unroll 4
    for (int l = 0; l < LDIM; ++l) s += attn_w[l] * enc[(size_t)l * HDIM + j];
    comb[j]        = emb[j];
    comb[HDIM + j] = s;
}

// PyTorch gate order i,f,g,o
__global__ void lstm_cell_kernel(const float* __restrict__ gates,
                                 const float* __restrict__ c_in,
                                 float* __restrict__ h_new,
                                 float* __restrict__ c_new)
{
    int i = blockIdx.x * blockDim.x + threadIdx.x;
    if (i >= HDIM) return;
    float ig = gates[i];
    float fg = gates[HDIM + i];
    float gg = gates[2 * HDIM + i];
    float og = gates[3 * HDIM + i];
    float si = 1.0f / (1.0f + expf(-ig));
    float sf = 1.0f / (1.0f + expf(-fg));
    float so = 1.0f / (1.0f + expf(-og));
    float cn = sf * c_in[i] + si * tanhf(gg);
    c_new[i] = cn;
    h_new[i] = so * tanhf(cn);
}

// online (max, sumexp) partials over the V logits
__global__ __launch_bounds__(256) void logits_partial_kernel(
    const float* __restrict__ logits, int n,
    float* __restrict__ pm, float* __restrict__ ps)
{
    __shared__ float shm[256], shs[256];
    int t = threadIdx.x;
    int gid = blockIdx.x * blockDim.x + t;
    int stride = gridDim.x * blockDim.x;
    float m = -INFINITY, s = 0.0f;
    for (int i = gid; i < n; i += stride) {
        float v = logits[i];
        if (v > m) { s = s * expf(m - v) + 1.0f; m = v; }
        else       { s += expf(v - m); }
    }
    shm[t] = m; shs[t] = s; __syncthreads();
    for (int st = 128; st > 0; st >>= 1) {
        if (t < st) {
            float m2 = shm[t + st], s2 = shs[t + st];
            float M = fmaxf(shm[t], m2);
            shs[t] = shs[t] * expf(shm[t] - M) + s2 * expf(m2 - M);
            shm[t] = M;
        }
        __syncthreads();
    }
    if (t == 0) { pm[blockIdx.x] = shm[0]; ps[blockIdx.x] = shs[0]; }
}

__global__ void logits_combine_kernel(const float* __restrict__ pm,
                                      const float* __restrict__ ps,
                                      float* __restrict__ stats)
{
    __shared__ float shm[128], shs[128];
    int t = threadIdx.x;
    shm[t] = pm[t]; shs[t] = ps[t]; __syncthreads();
    for (int st = 64; st > 0; st >>= 1) {
        if (t < st) {
            float m2 = shm[t + st], s2 = shs[t + st];
            float M = fmaxf(shm[t], m2);
            shs[t] = shs[t] * expf(shm[t] - M) + s2 * expf(m2 - M);
            shm[t] = M;
        }
        __syncthreads();
    }
    if (t == 0) { stats[0] = shm[0]; stats[1] = logf(shs[0]); }
}

__global__ void write_logp_kernel(const float* __restrict__ logits,
                                  const float* __restrict__ stats,
                                  float* __restrict__ out, int n)
{
    int i = blockIdx.x * blockDim.x + threadIdx.x;
    if (i >= n) return;
    out[i] = logits[i] - stats[0] - stats[1];
}

extern "C" void kernel_launch(void* const* d_in, const int* in_sizes, int n_in,
                              void* d_out, int out_size, void* d_ws, size_t ws_size,
                              hipStream_t stream)
{
    (void)in_sizes; (void)n_in; (void)out_size; (void)ws_size;
    const int*   x      = (const int*)  d_in[0];
    const float* enc    = (const float*)d_in[1];
    const float* h      = (const float*)d_in[2];
    const float* c      = (const float*)d_in[3];
    const float* emb_W  = (const float*)d_in[4];
    const float* attn_W = (const float*)d_in[5];
    const float* attn_b = (const float*)d_in[6];
    const float* comb_W = (const float*)d_in[7];
    const float* comb_b = (const float*)d_in[8];
    const float* W_ih   = (const float*)d_in[9];
    const float* W_hh   = (const float*)d_in[10];
    const float* b_ih   = (const float*)d_in[11];
    const float* b_hh   = (const float*)d_in[12];
    const float* out_W  = (const float*)d_in[13];
    const float* out_b  = (const float*)d_in[14];

    float* out = (float*)d_out;                // [V logp][L attn]
    float* ws  = (float*)d_ws;

    float* cat_he  = ws + OFF_CAT;
    float* alog    = ws + OFF_ALOG;
    float* aw      = ws + OFF_AW;
    float* comb    = ws + OFF_COMB;
    float* lstm_in = ws + OFF_LSTMIN;
    float* gates   = ws + OFF_GATES;
    float* h_new   = ws + OFF_HNEW;
    float* c_new   = ws + OFF_CNEW;
    float* logits  = ws + OFF_LOGITS;
    float* pm      = ws + OFF_PM;
    float* ps      = ws + OFF_PS;
    float* stats   = ws + OFF_STATS;

    // 1) [h0 || embedded]
    prep_cat_kernel<<<(2 * HDIM + 255) / 256, 256, 0, stream>>>(x, h, emb_W, cat_he);

    // 2) attn logits = cat_he @ attn_W.T + attn_b   (80 x 4096)
    wmma_gemv_kernel<<<1, 256, 2 * HDIM * sizeof(float), stream>>>(
        attn_W, cat_he, attn_b, alog, LDIM, 2 * HDIM, 0, 0);

    // 3) softmax over L, also writes attention output
    attn_softmax_kernel<<<1, 128, 0, stream>>>(alog, aw, out + VDIM);

    // 4) comb = [embedded || attn_w @ enc]
    attn_apply_kernel<<<(HDIM + 255) / 256, 256, 0, stream>>>(aw, enc, x, emb_W, comb);

    // 5) lstm_in = relu(comb @ comb_W.T + comb_b)   (2048 x 4096)
    wmma_gemv_kernel<<<(HDIM + 127) / 128, 256, 2 * HDIM * sizeof(float), stream>>>(
        comb_W, comb, comb_b, lstm_in, HDIM, 2 * HDIM, 0, 1);

    // 6) gates = lstm_in @ W_ih.T + b_ih            (8192 x 2048)
    wmma_gemv_kernel<<<(4 * HDIM + 127) / 128, 256, HDIM * sizeof(float), stream>>>(
        W_ih, lstm_in, b_ih, gates, 4 * HDIM, HDIM, 0, 0);

    // 7) gates += h0 @ W_hh.T + b_hh                (8192 x 2048, accumulate)
    wmma_gemv_kernel<<<(4 * HDIM + 127) / 128, 256, HDIM * sizeof(float), stream>>>(
        W_hh, h, b_hh, gates, 4 * HDIM, HDIM, 1, 0);

    // 8) LSTM elementwise
    lstm_cell_kernel<<<(HDIM + 255) / 256, 256, 0, stream>>>(gates, c, h_new, c_new);

    // 9) logits = h_new @ out_W.T + out_b           (50257 x 2048) — dominant, NT streamed
    wmma_gemv_kernel<<<(VDIM + 127) / 128, 256, HDIM * sizeof(float), stream>>>(
        out_W, h_new, out_b, logits, VDIM, HDIM, 0, 0);

    // 10-12) log-softmax over V
    logits_partial_kernel<<<128, 256, 0, stream>>>(logits, VDIM, pm, ps);
    logits_combine_kernel<<<1, 128, 0, stream>>>(pm, ps, stats);
    write_logp_kernel<<<(VDIM + 255) / 256, 256, 0, stream>>>(logits, stats, out, VDIM);
}